// Pts3D_Regr3D_CD_V4_84739704750726
// MI455X (gfx1250) — compile-verified
//
#include <hip/hip_runtime.h>
#include <hip/hip_bf16.h>

// Problem constants (match reference)
constexpr int BB = 4;
constexpr int NP = 8192;
constexpr int NG = 8192;

constexpr int TILES  = 4;            // row tiles (16 rows each) per wave
constexpr int WAVES  = 8;            // waves per workgroup
constexpr int ROWS_WG = WAVES * TILES * 16;  // 512 pred rows per workgroup
constexpr int SEGS   = 4;            // column-sweep segments (parallelism)
constexpr int SEG_COLS = NG / SEGS;  // 2048 gt columns per segment

typedef __attribute__((ext_vector_type(2))) float v2f;
typedef __attribute__((ext_vector_type(8))) float v8f;

#define INF_IBITS 0x7F800000   // +inf bit pattern (large positive as int)

// ---------------------------------------------------------------------------
// Kernel 1: precompute
//   bprime[b,m]  = {-2gx, -2gy, -2gz, |g|^2}   (float4 per gt point)
//   a2[b,n]      = |p|^2                        (per pred point)
//   colmin[b,m]  = +inf bits,  rowmin[b,n] = +inf bits
// ---------------------------------------------------------------------------
__global__ void cd_prep_kernel(const float* __restrict__ pr,
                               const float* __restrict__ gt,
                               float* __restrict__ bprime,
                               float* __restrict__ a2,
                               int* __restrict__ rowmin,
                               int* __restrict__ colmin) {
  int i = blockIdx.x * blockDim.x + threadIdx.x;
  if (i < BB * NG) {
    float x = gt[3 * i + 0];
    float y = gt[3 * i + 1];
    float z = gt[3 * i + 2];
    float g2 = x * x + y * y + z * z;
    reinterpret_cast<float4*>(bprime)[i] =
        make_float4(-2.0f * x, -2.0f * y, -2.0f * z, g2);
    colmin[i] = INF_IBITS;
  }
  if (i < BB * NP) {
    float x = pr[3 * i + 0];
    float y = pr[3 * i + 1];
    float z = pr[3 * i + 2];
    a2[i] = x * x + y * y + z * z;
    rowmin[i] = INF_IBITS;
  }
}

// ---------------------------------------------------------------------------
// Kernel 2: pairwise distance tiles via V_WMMA_F32_16X16X4_F32.
//
// D = A' x B' + C  with  A' = [px,py,pz,1], B' = [-2gx,-2gy,-2gz,|g|^2],
// C = |p|^2 broadcast per row  =>  D[n,m] = |p-g|^2 straight from the
// matrix pipe, zero VALU arithmetic.
//
// Block = 256 threads = 8 waves; each wave owns 4 row tiles (64 pred rows)
// and sweeps one column segment (2048 gt points = 128 column tiles); the B
// fragment is loaded once and reused by 4 back-to-back WMMAs.
//
// All mins operate on RAW float bit patterns with SIGNED integer min
// (v_min_i32 / v_min3_i32): correct float ordering for all d2 >= 0 and for
// sign crossings; no clamps, no canonicalization ops.
// gt-validity mask rides on EXEC: row-min updates are wrapped in `if (vm)`
// (lane-uniform over a lane's 32 elements), so invalid columns simply never
// commit. WMMAs execute before the divergent region (EXEC all-1s).
// Row mins: registers -> shuffle smin reduce -> global signed atomicMin.
// Col mins: LDS int array (ds_min_i32, all 32 lanes) -> global atomicMin.
// ---------------------------------------------------------------------------
__global__ void __launch_bounds__(256)
cd_wmma_kernel(const float* __restrict__ pr,
               const unsigned char* __restrict__ gvalid,
               const float* __restrict__ bprime,
               const float* __restrict__ a2,
               int* __restrict__ rowmin,
               int* __restrict__ colmin) {
  __shared__ int ldsCol[SEG_COLS];  // 8 KB

  const int tid  = threadIdx.x;
  const int lane = tid & 31;
  const int wave = tid >> 5;
  const int half = lane >> 4;   // 0: rows 0-7 / K=0,1 ; 1: rows 8-15 / K=2,3
  const int l15  = lane & 15;

  // init LDS column mins
  for (int i = tid; i < SEG_COLS; i += 256) ldsCol[i] = INF_IBITS;
  __syncthreads();

  // grid decomposition: blockIdx.x = (b * (NP/ROWS_WG) + rblk) * SEGS + seg
  const int seg  = blockIdx.x % SEGS;
  const int tmp  = blockIdx.x / SEGS;
  const int rblk = tmp % (NP / ROWS_WG);
  const int b    = tmp / (NP / ROWS_WG);

  const int waveRow = rblk * ROWS_WG + wave * (TILES * 16);
  const size_t bBaseP = (size_t)b * NP;
  const size_t bBaseG = (size_t)b * NG;

  // ---- A fragments (16x4 each): lane<16 -> (px,py); lane>=16 -> (pz,1)
  v2f av[TILES];
#pragma unroll
  for (int t = 0; t < TILES; ++t) {
    const float* ap = pr + (bBaseP + waveRow + t * 16 + l15) * 3;
    if (half == 0) { av[t][0] = ap[0]; av[t][1] = ap[1]; }
    else           { av[t][0] = ap[2]; av[t][1] = 1.0f;  }
  }

  // ---- WMMA C operand per tile: |p|^2 for the 8 rows this lane accumulates
  v8f a2c[TILES];
#pragma unroll
  for (int t = 0; t < TILES; ++t) {
    const float* p = a2 + bBaseP + waveRow + t * 16 + half * 8;
#pragma unroll
    for (int r = 0; r < 8; ++r) a2c[t][r] = p[r];
  }

  // ---- running masked row mins (raw d2 bits, signed-int ordering)
  int rmin[TILES][8];
#pragma unroll
  for (int t = 0; t < TILES; ++t)
#pragma unroll
    for (int r = 0; r < 8; ++r) rmin[t][r] = INF_IBITS;

  const int jt0 = seg * (SEG_COLS / 16);
  const int jt1 = jt0 + (SEG_COLS / 16);

  for (int jt = jt0; jt < jt1; ++jt) {
    const int m = jt * 16 + l15;               // this lane's gt column
    const size_t mIdx = bBaseG + m;

    // B fragment (4x16): lane<16 -> K=0,1 comps; lane>=16 -> K=2,3 comps
    const float* bp = bprime + mIdx * 4 + half * 2;
    v2f bv; bv[0] = bp[0]; bv[1] = bp[1];

    if (jt + 8 < jt1)
      __builtin_prefetch(bprime + (mIdx + 8 * 16) * 4, 0, 3);

    const unsigned char vm = gvalid[mIdx];  // lane-uniform over this lane's rows

    // ---- 4 d2 tiles straight from the matrix pipe (EXEC all-1s here)
    v8f d[TILES];
#pragma unroll
    for (int t = 0; t < TILES; ++t)
      d[t] = __builtin_amdgcn_wmma_f32_16x16x4_f32(
          /*neg_a=*/false, av[t], /*neg_b=*/false, bv,
          /*c_mod=*/(short)0, a2c[t], /*reuse_a=*/false, /*reuse_b=*/false);

    // ---- unmasked column min over all 32 rows (signed bits, min3 tree)
    int colu = INF_IBITS;
#pragma unroll
    for (int t = 0; t < TILES; ++t)
#pragma unroll
      for (int r = 0; r < 8; ++r)
        colu = min(colu, __float_as_int(d[t][r]));
    // both halves hit the same slot (lanes l and l+16 share column m)
    atomicMin(&ldsCol[m - seg * SEG_COLS], colu);

    // ---- masked row mins: invalid lanes masked out via EXEC
    if (vm) {
#pragma unroll
      for (int t = 0; t < TILES; ++t)
#pragma unroll
        for (int r = 0; r < 8; ++r)
          rmin[t][r] = min(rmin[t][r], __float_as_int(d[t][r]));
    }
  }

  // ---- reduce row mins across the 16 lanes of each half, flush to global
#pragma unroll
  for (int t = 0; t < TILES; ++t)
#pragma unroll
    for (int r = 0; r < 8; ++r) {
#pragma unroll
      for (int off = 1; off < 16; off <<= 1)
        rmin[t][r] = min(rmin[t][r], __shfl_xor(rmin[t][r], off, 32));
    }
  if (l15 == 0) {
#pragma unroll
    for (int t = 0; t < TILES; ++t) {
      int* rout = rowmin + bBaseP + waveRow + t * 16 + half * 8;
#pragma unroll
      for (int r = 0; r < 8; ++r)
        atomicMin(&rout[r], rmin[t][r]);
    }
  }

  // ---- flush LDS column mins to global
  __syncthreads();
  for (int i = tid; i < SEG_COLS; i += 256)
    atomicMin(&colmin[bBaseG + seg * SEG_COLS + i], ldsCol[i]);
}

// ---------------------------------------------------------------------------
// Kernel 3: final reduction -> scalar 2*(mean_b acc_b + mean_b com_b)
// ---------------------------------------------------------------------------
__device__ __forceinline__ float block_reduce_sum(float v, float* sm) {
  int t = threadIdx.x;
  sm[t] = v;
  __syncthreads();
  for (int s = 128; s > 0; s >>= 1) {
    if (t < s) sm[t] += sm[t + s];
    __syncthreads();
  }
  float r = sm[0];
  __syncthreads();
  return r;
}

__global__ void cd_reduce_kernel(const int* __restrict__ rowmin,
                                 const int* __restrict__ colmin,
                                 const unsigned char* __restrict__ gvalid,
                                 float* __restrict__ out) {
  __shared__ float sm[256];
  float accTotal = 0.0f, comTotal = 0.0f;
  for (int b = 0; b < BB; ++b) {
    // accuracy: mean_n rowmin  (rowmin holds raw bits of masked min d2)
    float s = 0.0f;
    for (int n = threadIdx.x; n < NP; n += 256)
      s += __int_as_float(rowmin[(size_t)b * NP + n]);
    float accSum = block_reduce_sum(s, sm);
    // completeness: sum over valid m of colmin / count
    float cs = 0.0f, cc = 0.0f;
    for (int m = threadIdx.x; m < NG; m += 256) {
      size_t i = (size_t)b * NG + m;
      if (gvalid[i]) {
        cs += __int_as_float(colmin[i]);
        cc += 1.0f;
      }
    }
    float comSum = block_reduce_sum(cs, sm);
    float cntSum = block_reduce_sum(cc, sm);
    accTotal += accSum / (float)NP;
    comTotal += comSum / fmaxf(cntSum, 1.0f);
  }
  if (threadIdx.x == 0) {
    float loss_acc = accTotal / (float)BB;
    float loss_com = comTotal / (float)BB;
    out[0] = 2.0f * (loss_acc + loss_com);  // acc + com + cd
  }
}

// ---------------------------------------------------------------------------
extern "C" void kernel_launch(void* const* d_in, const int* in_sizes, int n_in,
                              void* d_out, int out_size, void* d_ws, size_t ws_size,
                              hipStream_t stream) {
  const float* pr         = (const float*)d_in[0];           // [B,NP,3] f32
  const float* gt         = (const float*)d_in[1];           // [B,NG,3] f32
  const unsigned char* gv = (const unsigned char*)d_in[2];   // [B,NG] bool (1B)
  float* out              = (float*)d_out;

  // workspace layout: bprime | a2 | rowmin | colmin
  float* bprime = (float*)d_ws;                  // B*NG*4 floats
  float* a2     = bprime + (size_t)BB * NG * 4;  // B*NP floats
  int* rowmin   = (int*)(a2 + (size_t)BB * NP);  // B*NP ints
  int* colmin   = rowmin + (size_t)BB * NP;      // B*NG ints

  // 1) precompute + min-buffer init
  {
    int total = BB * ((NP > NG) ? NP : NG);
    cd_prep_kernel<<<(total + 255) / 256, 256, 0, stream>>>(
        pr, gt, bprime, a2, rowmin, colmin);
  }
  // 2) WMMA pairwise tile sweep
  {
    dim3 grid(BB * (NP / ROWS_WG) * SEGS);  // 4 * 16 * 4 = 256 workgroups
    cd_wmma_kernel<<<grid, 256, 0, stream>>>(pr, gv, bprime, a2, rowmin, colmin);
  }
  // 3) scalar reduction
  cd_reduce_kernel<<<1, 256, 0, stream>>>(rowmin, colmin, gv, out);
}